// MelPCENTransform_15728170237987
// MI455X (gfx1250) — compile-verified
//
#include <hip/hip_runtime.h>
#include <math.h>

typedef __attribute__((ext_vector_type(2))) float v2f;
typedef __attribute__((ext_vector_type(8))) float v8f;

#define SR       32000
#define NFFT     2048
#define HOP      512
#define NMELS    128
#define NB       64
#define LSIG     160000
#define TFRAMES  313          // 1 + 160000/512
#define TILE_T   16
#define NTILES   20           // ceil(313/16)
#define NC       1024         // complex FFT length (real FFT of 2048 via packing)
#define KP       1028         // K padded to multiple of 4
#define PAD      1024
#define NTHREADS 256          // 8 wave32 per workgroup

// LDS layout (dynamic shared):
//   zbuf : float2[16*1024]  = 131072 B   (FFT working buffer)
//   pbuf : float [16*1028]  =  65792 B   (power spectrum, zero-padded K)
//   fp   : float [130]      (mel breakpoint freqs)
//   invd : float [129]      (1 / diff(fp))
#define ZBUF_BYTES (TILE_T * NC * 8)
#define PBUF_BYTES (TILE_T * KP * 4)
#define SH_BYTES   (ZBUF_BYTES + PBUF_BYTES + 130 * 4 + 129 * 4)

__device__ __forceinline__ int reflect_idx(int i) {
    i = (i < 0) ? -i : i;
    i = (i >= LSIG) ? (2 * LSIG - 2 - i) : i;
    return i;
}

__global__ void __launch_bounds__(NTHREADS)
melspec_kernel(const float* __restrict__ wav, float* __restrict__ out) {
    extern __shared__ char smem[];
    float2* zbuf = (float2*)smem;
    float*  pbuf = (float*)(smem + ZBUF_BYTES);
    float*  fp   = (float*)(smem + ZBUF_BYTES + PBUF_BYTES);
    float*  invd = fp + 130;

    const int tid  = threadIdx.x;
    const int tile = blockIdx.x;      // frame tile
    const int bb   = blockIdx.y;      // batch
    const int t0   = tile * TILE_T;

    // ---- mel breakpoints (HTK mel scale), computed in LDS ----
    if (tid < 130) {
        float mlo = 2595.0f * log10f(1.0f + 20.0f / 700.0f);
        float mhi = 2595.0f * log10f(1.0f + 16000.0f / 700.0f);
        float mm  = mlo + (mhi - mlo) * ((float)tid / 129.0f);
        fp[tid] = 700.0f * (powf(10.0f, mm / 2595.0f) - 1.0f);
    }
    __syncthreads();
    if (tid < 129) invd[tid] = 1.0f / (fp[tid + 1] - fp[tid]);

    // ---- load + window + bit-reversed scatter (pack even/odd into complex) ----
    const float* wb = wav + (size_t)bb * LSIG;
    for (int idx = tid; idx < TILE_T * NC; idx += NTHREADS) {
        int fr = idx >> 10;
        int n  = idx & (NC - 1);
        int start = (t0 + fr) * HOP - PAD;
        float s0 = wb[reflect_idx(start + 2 * n)];
        float s1 = wb[reflect_idx(start + 2 * n + 1)];
        float w0 = 0.5f - 0.5f * cospif((float)(2 * n)     * (1.0f / 1024.0f));
        float w1 = 0.5f - 0.5f * cospif((float)(2 * n + 1) * (1.0f / 1024.0f));
        int br = (int)(__brev((unsigned)n) >> 22);   // 10-bit reverse
        zbuf[fr * NC + br] = make_float2(s0 * w0, s1 * w1);
    }

    // ---- 10-stage radix-2 DIT FFT (forward), all frames cooperatively ----
    for (int s = 0; s < 10; ++s) {
        __syncthreads();
        int half = 1 << s;
        for (int i = tid; i < TILE_T * (NC / 2); i += NTHREADS) {
            int fr = i >> 9;
            int j  = i & 511;
            int p  = j & (half - 1);
            int g  = j >> s;
            int i0 = fr * NC + (g << (s + 1)) + p;
            int i1 = i0 + half;
            float sn, cs;
            sincospif(-(float)p / (float)half, &sn, &cs);   // e^{-i*pi*p/half}
            float2 z0 = zbuf[i0];
            float2 z1 = zbuf[i1];
            float tr = cs * z1.x - sn * z1.y;
            float ti = cs * z1.y + sn * z1.x;
            zbuf[i0] = make_float2(z0.x + tr, z0.y + ti);
            zbuf[i1] = make_float2(z0.x - tr, z0.y - ti);
        }
    }
    __syncthreads();

    // ---- real-FFT untwiddle -> power spectrum (K padded with zeros) ----
    for (int idx = tid; idx < TILE_T * KP; idx += NTHREADS) {
        int fr = idx / KP;
        int k  = idx - fr * KP;
        float pw = 0.0f;
        if (k <= NC) {
            float2 Za = zbuf[fr * NC + (k & (NC - 1))];
            float2 Zb = zbuf[fr * NC + ((NC - k) & (NC - 1))];
            float er = 0.5f * (Za.x + Zb.x);
            float ei = 0.5f * (Za.y - Zb.y);
            float dr = 0.5f * (Za.x - Zb.x);
            float di = 0.5f * (Za.y + Zb.y);
            float orr =  di;     // Zo = -i/2 * (Za - conj(Zb))
            float oii = -dr;
            float sn, cs;
            sincospif(-(float)k * (1.0f / 1024.0f), &sn, &cs);
            float Xr = er + cs * orr - sn * oii;
            float Xi = ei + cs * oii + sn * orr;
            pw = Xr * Xr + Xi * Xi;
        }
        pbuf[idx] = pw;
    }
    __syncthreads();

    // ---- mel projection: [16 x KP] * [KP x 128] via V_WMMA_F32_16X16X4_F32 ----
    // 8 waves, each owns one 16-mel N-tile; K consumed in chunks of 4.
    const int lane = tid & 31;
    const int wv   = tid >> 5;        // wave 0..7 -> mels [16*wv, 16*wv+16)
    const int hi   = lane >> 4;       // lane half selects K sub-pair (+2)
    const int l15  = lane & 15;       // A row (frame) == B col (mel within tile)
    const int m    = wv * 16 + l15;   // this lane's mel column

    // loop-invariant triangular-filter params for mel m
    const float fm0 = fp[m];
    const float fm2 = fp[m + 2];
    const float id0 = invd[m];
    const float id1 = invd[m + 1];

    v8f acc = {};
    for (int kc = 0; kc < KP / 4; ++kc) {
        int kk = kc * 4 + 2 * hi;
        // A fragment: lane l, VGPR j holds power[row=l&15][4kc + j + 2*(l>=16)]
        v2f a = *(const v2f*)(pbuf + l15 * KP + kk);
        // B fragment: fb[K][m], same K mapping as A
        float fq0 = (float)kk * 15.625f;
        float fq1 = fq0 + 15.625f;
        v2f b;
        b.x = fmaxf(0.0f, fminf((fq0 - fm0) * id0, (fm2 - fq0) * id1));
        b.y = fmaxf(0.0f, fminf((fq1 - fm0) * id0, (fm2 - fq1) * id1));
        acc = __builtin_amdgcn_wmma_f32_16x16x4_f32(false, a, false, b, (short)0, acc, false, false);
    }

    // ---- scatter D to out[b][mel][t]  (VGPR i -> rows i and i+8) ----
    float* orow = out + ((size_t)bb * NMELS + m) * TFRAMES;
    #pragma unroll
    for (int i = 0; i < 8; ++i) {
        int t = t0 + i + 8 * hi;
        if (t < TFRAMES) orow[t] = acc[i];
    }
}

// PCEN: sequential scan over t, one thread per (batch, mel), in-place on out.
__global__ void __launch_bounds__(128)
pcen_kernel(float* __restrict__ out) {
    const int bb = blockIdx.x;
    const int m  = threadIdx.x;
    float* row = out + ((size_t)bb * NMELS + m) * TFRAMES;
    const float S = 0.025f, ALPHA = 0.98f, EPS = 1e-6f;
    const float sqrt_delta = 1.41421356237309515f;   // sqrt(2) = DELTA^R
    float mst = 0.0f;
    for (int t = 0; t < TFRAMES; ++t) {
        float x = row[t];
        mst = (1.0f - S) * mst + S * x;
        float denom = powf(EPS + mst, ALPHA);
        row[t] = sqrtf(x / denom + 2.0f) - sqrt_delta;
    }
}

extern "C" void kernel_launch(void* const* d_in, const int* in_sizes, int n_in,
                              void* d_out, int out_size, void* d_ws, size_t ws_size,
                              hipStream_t stream) {
    (void)in_sizes; (void)n_in; (void)d_ws; (void)ws_size; (void)out_size;
    const float* wav = (const float*)d_in[0];
    float* out = (float*)d_out;

    // allow >64KB dynamic LDS (CDNA5 WGP has 320KB); idempotent, deterministic
    (void)hipFuncSetAttribute((const void*)melspec_kernel,
                              hipFuncAttributeMaxDynamicSharedMemorySize, SH_BYTES);

    dim3 grid(NTILES, NB);
    melspec_kernel<<<grid, NTHREADS, SH_BYTES, stream>>>(wav, out);
    pcen_kernel<<<NB, NMELS, 0, stream>>>(out);
}